// Zernike_62328565399852
// MI455X (gfx1250) — compile-verified
//
#include <hip/hip_runtime.h>
#include <hip/hip_bf16.h>

// Problem constants (match reference setup_inputs / get_zernike_combination)
#define NB 4
#define NA 192
#define NM 40
#define NT 780
#define NK 18     // number of (n1,n2,l) combos for N_MAX=5, L_MAX=3, DIAG=False
#define NE 3      // len(ELEMENTS)

typedef float v2f __attribute__((ext_vector_type(2)));
typedef float v8f __attribute__((ext_vector_type(8)));

// Only probe for the builtin in the DEVICE pass; host pass has no amdgcn builtins.
#if defined(__HIP_DEVICE_COMPILE__)
#if !__has_builtin(__builtin_amdgcn_wmma_f32_16x16x4_f32)
#error "missing __builtin_amdgcn_wmma_f32_16x16x4_f32 on device toolchain"
#endif
#endif

// cut_zernike(5,2)[n1[k], l[k]] : power-basis coeffs of R_nl(x)*(1-x)^2, D=8.
// Rows follow combo order: l=0:(0,0)(0,2)(0,4)(2,2)(2,4)(4,4)
//                          l=1:(1,1)(1,3)(1,5)(3,3)(3,5)(5,5)
//                          l=2:(2,2)(2,4)(4,4)   l=3:(3,3)(3,5)(5,5)
__constant__ float CAC[NK][8] = {
  { 1,-2, 1, 0, 0,  0,  0, 0},   // zc[0,0]
  { 1,-2, 1, 0, 0,  0,  0, 0},   // zc[0,0]
  { 1,-2, 1, 0, 0,  0,  0, 0},   // zc[0,0]
  {-1, 2, 1,-4, 2,  0,  0, 0},   // zc[2,0]
  {-1, 2, 1,-4, 2,  0,  0, 0},   // zc[2,0]
  { 1,-2,-5,12, 0,-12,  6, 0},   // zc[4,0]
  { 0, 1,-2, 1, 0,  0,  0, 0},   // zc[1,1]
  { 0, 1,-2, 1, 0,  0,  0, 0},   // zc[1,1]
  { 0, 1,-2, 1, 0,  0,  0, 0},   // zc[1,1]
  { 0,-2, 4, 1,-6,  3,  0, 0},   // zc[3,1]
  { 0,-2, 4, 1,-6,  3,  0, 0},   // zc[3,1]
  { 0, 3,-6,-9,24, -2,-20,10},   // zc[5,1]
  { 0, 0, 1,-2, 1,  0,  0, 0},   // zc[2,2]
  { 0, 0, 1,-2, 1,  0,  0, 0},   // zc[2,2]
  { 0, 0,-3, 6, 1, -8,  4, 0},   // zc[4,2]
  { 0, 0, 0, 1,-2,  1,  0, 0},   // zc[3,3]
  { 0, 0, 0, 1,-2,  1,  0, 0},   // zc[3,3]
  { 0, 0, 0,-4, 8,  1,-10, 5},   // zc[5,3]
};
__constant__ float CBC[NK][8] = {
  { 1,-2, 1, 0, 0,  0,  0, 0},   // zc[0,0]
  {-1, 2, 1,-4, 2,  0,  0, 0},   // zc[2,0]
  { 1,-2,-5,12, 0,-12,  6, 0},   // zc[4,0]
  {-1, 2, 1,-4, 2,  0,  0, 0},   // zc[2,0]
  { 1,-2,-5,12, 0,-12,  6, 0},   // zc[4,0]
  { 1,-2,-5,12, 0,-12,  6, 0},   // zc[4,0]
  { 0, 1,-2, 1, 0,  0,  0, 0},   // zc[1,1]
  { 0,-2, 4, 1,-6,  3,  0, 0},   // zc[3,1]
  { 0, 3,-6,-9,24, -2,-20,10},   // zc[5,1]
  { 0,-2, 4, 1,-6,  3,  0, 0},   // zc[3,1]
  { 0, 3,-6,-9,24, -2,-20,10},   // zc[5,1]
  { 0, 3,-6,-9,24, -2,-20,10},   // zc[5,1]
  { 0, 0, 1,-2, 1,  0,  0, 0},   // zc[2,2]
  { 0, 0,-3, 6, 1, -8,  4, 0},   // zc[4,2]
  { 0, 0,-3, 6, 1, -8,  4, 0},   // zc[4,2]
  { 0, 0, 0, 1,-2,  1,  0, 0},   // zc[3,3]
  { 0, 0, 0,-4, 8,  1,-10, 5},   // zc[5,3]
  { 0, 0, 0,-4, 8,  1,-10, 5},   // zc[5,3]
};
// legendre(3)[l[k]] power-basis coeffs, Dl=4
__constant__ float CLC[NK][4] = {
  {1,0,0,0},{1,0,0,0},{1,0,0,0},{1,0,0,0},{1,0,0,0},{1,0,0,0},           // P0
  {0,1,0,0},{0,1,0,0},{0,1,0,0},{0,1,0,0},{0,1,0,0},{0,1,0,0},           // P1
  {-0.5f,0,1.5f,0},{-0.5f,0,1.5f,0},{-0.5f,0,1.5f,0},                    // P2
  {0,-1.5f,0,2.5f},{0,-1.5f,0,2.5f},{0,-1.5f,0,2.5f},                    // P3
};

static __device__ __forceinline__ v8f wmma4(v2f a, v2f b, v8f c) {
  // D(16x16,f32) = A(16x4,f32) x B(4x16,f32) + C
  return __builtin_amdgcn_wmma_f32_16x16x4_f32(
      /*neg_a=*/false, a, /*neg_b=*/false, b,
      /*c_mod=*/(short)0, c, /*reuse_a=*/false, /*reuse_b=*/false);
}

__global__ __launch_bounds__(128) void zernike_wmma_kernel(
    const float* __restrict__ pos,   // [B,N,3]
    const float* __restrict__ cell,  // [B,3,3]
    const int*   __restrict__ nbr,   // [B,N,M]
    const int*   __restrict__ nbrj,  // [B,N,T]
    const int*   __restrict__ nbrk,  // [B,N,T]
    const float* __restrict__ msk,   // [B,N,M]
    const float* __restrict__ mskt,  // [B,N,T]
    const float* __restrict__ off,   // [B,N,M,3]
    const float* __restrict__ offj,  // [B,N,T,3]
    const float* __restrict__ offk,  // [B,N,T,3]
    const int*   __restrict__ anum,  // [B,N]
    float*       __restrict__ out)   // [B,N,E*K]
{
  const int bn   = blockIdx.x;       // atom-center index b*N+n
  const int b    = bn / NA;
  const int lane = threadIdx.x & 31; // wave32 lane
  const int wv   = threadIdx.x >> 5; // wave id within block (0..3)
  const int nn   = lane & 15;        // column (combo) within N-tile
  const int hi   = lane >> 4;        // half: K-element offset 0 or 2
  const int ko   = hi << 1;

  __shared__ float red[4][NK];

  // cell matrix (rows c, cols d): cart_d = sum_c off_c * cell[c][d]
  const float* C = cell + b * 9;
  const float c00=C[0], c01=C[1], c02=C[2];
  const float c10=C[3], c11=C[4], c12=C[5];
  const float c20=C[6], c21=C[7], c22=C[8];

  // Constant B operands (K=4 x N=16 tiles): lane nn holds column 16*t+nn,
  // VGPR0 = row 4s+ko, VGPR1 = row 4s+ko+1.
  v2f BA[2][2], BB[2][2], BL[2];
#pragma unroll
  for (int t = 0; t < 2; ++t) {
    const int cidx = t * 16 + nn;
    const bool vld = cidx < NK;
    const int  ci  = vld ? cidx : 0;
#pragma unroll
    for (int s = 0; s < 2; ++s) {
      v2f ba; ba.x = vld ? CAC[ci][4*s+ko]   : 0.f;
              ba.y = vld ? CAC[ci][4*s+ko+1] : 0.f;
      v2f bb; bb.x = vld ? CBC[ci][4*s+ko]   : 0.f;
              bb.y = vld ? CBC[ci][4*s+ko+1] : 0.f;
      BA[t][s] = ba; BB[t][s] = bb;
    }
    v2f bl; bl.x = vld ? CLC[ci][ko]   : 0.f;
            bl.y = vld ? CLC[ci][ko+1] : 0.f;
    BL[t] = bl;
  }

  const v8f vzero = {0.f,0.f,0.f,0.f,0.f,0.f,0.f,0.f};
  v8f acc[2] = {vzero, vzero};

  const float pix = pos[bn*3+0], piy = pos[bn*3+1], piz = pos[bn*3+2];

  // ---------------- pair term: f1[k] = sum_m 2*RA*RB * zij^2 (P_l(1)=1) ----
  const int MCH = (NM + 15) / 16;
  for (int ch = wv; ch < MCH; ch += 4) {
    const int   m  = ch*16 + nn;
    const float pv = (m < NM) ? 1.f : 0.f;
    const int   mc = (m < NM) ? m : (NM-1);
    const int base = bn*NM + mc;
    const int   jn = nbr[base];
    const float w  = msk[base] * pv;
    const float o0 = off[base*3+0], o1 = off[base*3+1], o2 = off[base*3+2];
    const int   jb = b*NA + jn;
    const float vx = pos[jb*3+0] - pix + o0*c00 + o1*c10 + o2*c20;
    const float vy = pos[jb*3+1] - piy + o0*c01 + o1*c11 + o2*c21;
    const float vz = pos[jb*3+2] - piz + o0*c02 + o1*c12 + o2*c22;
    const float d2 = vx*vx + vy*vy + vz*vz;
    const float r  = sqrtf(w > 0.f ? d2 : 1.f);   // safe sqrt (masked -> 1)
    const float x  = r * 0.2f;                    // r / CUTOFF
    const float zij = (float)anum[jb] * 0.125f;   // z_ratio (max element = 8)
    const float wz  = w * zij;                    // mask*zij folded into A row
    const float x2 = x*x, x4 = x2*x2;
    const float a0 = hi ? wz*x2 : wz;             // zij * x^{ko}
    const v2f A0 = {a0,    a0*x};                 // powers ko, ko+1
    const v2f A1 = {a0*x4, a0*x*x4};              // powers ko+4, ko+5
#pragma unroll
    for (int t = 0; t < 2; ++t) {
      v8f DA = wmma4(A1, BA[t][1], wmma4(A0, BA[t][0], vzero)); // pw.cA
      v8f DB = wmma4(A1, BB[t][1], wmma4(A0, BB[t][0], vzero)); // pw.cB
      acc[t] += (DA * DB) * 2.0f;   // zij^2, mask^2==mask come along for free
    }
  }

  // ---------------- triple term ------------------------------------------
  const int TCH = (NT + 15) / 16;
  for (int ch = wv; ch < TCH; ch += 4) {
    const int   tt = ch*16 + nn;
    const float pv = (tt < NT) ? 1.f : 0.f;
    const int   tc = (tt < NT) ? tt : (NT-1);
    const int base = bn*NT + tc;
    const int   jn = nbrj[base], kn = nbrk[base];
    const float w  = mskt[base] * pv;
    const float oj0 = offj[base*3+0], oj1 = offj[base*3+1], oj2 = offj[base*3+2];
    const float ok0 = offk[base*3+0], ok1 = offk[base*3+1], ok2 = offk[base*3+2];
    const int jb = b*NA + jn, kb = b*NA + kn;
    const float Rjx = pos[jb*3+0] + oj0*c00 + oj1*c10 + oj2*c20 - pix;
    const float Rjy = pos[jb*3+1] + oj0*c01 + oj1*c11 + oj2*c21 - piy;
    const float Rjz = pos[jb*3+2] + oj0*c02 + oj1*c12 + oj2*c22 - piz;
    const float Rkx = pos[kb*3+0] + ok0*c00 + ok1*c10 + ok2*c20 - pix;
    const float Rky = pos[kb*3+1] + ok0*c01 + ok1*c11 + ok2*c21 - piy;
    const float Rkz = pos[kb*3+2] + ok0*c02 + ok1*c12 + ok2*c22 - piz;
    const float d2j  = Rjx*Rjx + Rjy*Rjy + Rjz*Rjz;
    const float d2k  = Rkx*Rkx + Rky*Rky + Rkz*Rkz;
    const float dxx = Rjx-Rkx, dyy = Rjy-Rky, dzz = Rjz-Rkz;
    const float d2jk = dxx*dxx + dyy*dyy + dzz*dzz;
    const bool live = w > 0.f;
    const float rj   = sqrtf(live ? d2j : 1.f);
    const float rk   = sqrtf(live ? d2k : 1.f);
    const float rjk2 = live ? d2jk : 1.f;
    const float cosv = live ? (rj*rj + rk*rk - rjk2) / (2.f*rj*rk) : 0.f;
    const float zj = (float)anum[jb] * 0.125f;
    const float zk = (float)anum[kb] * 0.125f;
    const float xj = rj*0.2f, xk = rk*0.2f;
    const float xj2 = xj*xj, xj4 = xj2*xj2;
    const float xk2 = xk*xk, xk4 = xk2*xk2;
    const float aj0 = hi ? zj*xj2 : zj;       // zj * xj^{ko}
    const v2f Aj0 = {aj0,     aj0*xj};
    const v2f Aj1 = {aj0*xj4, aj0*xj*xj4};
    const float ak0 = hi ? zk*xk2 : zk;       // zk * xk^{ko}
    const v2f Ak0 = {ak0,     ak0*xk};
    const v2f Ak1 = {ak0*xk4, ak0*xk*xk4};
    const float cc2 = cosv*cosv;
    const float ac0 = hi ? w*cc2 : w;         // mask * cos^{ko}
    const v2f Ac = {ac0, ac0*cosv};
#pragma unroll
    for (int t = 0; t < 2; ++t) {
      v8f D1 = wmma4(Aj1, BA[t][1], wmma4(Aj0, BA[t][0], vzero)); // zj*pwj . cA
      v8f D2 = wmma4(Ak1, BB[t][1], wmma4(Ak0, BB[t][0], vzero)); // zk*pwk . cB
      v8f pr = D1 * D2;
      v8f D3 = wmma4(Ak1, BA[t][1], wmma4(Ak0, BA[t][0], vzero)); // zk*pwk . cA
      v8f D4 = wmma4(Aj1, BB[t][1], wmma4(Aj0, BB[t][0], vzero)); // zj*pwj . cB
      pr += D3 * D4;                                              // rad1+rad2 (x zjk)
      v8f D5 = wmma4(Ac, BL[t], vzero);                           // mask*pwc . cL
      acc[t] += pr * D5;
    }
  }

  // ------- reduce over the 16 M-rows of each accumulator tile -------------
  float part0 = 0.f, part1 = 0.f;
#pragma unroll
  for (int i = 0; i < 8; ++i) { part0 += acc[0][i]; part1 += acc[1][i]; }
  // rows 0..7 live in lanes 0..15, rows 8..15 in lanes 16..31 (same column)
  const float f0 = part0 + __shfl_xor(part0, 16, 32);
  const float f1 = part1 + __shfl_xor(part1, 16, 32);

  if (hi == 0) {
    red[wv][nn] = f0;               // combos 0..15
    if (nn < NK - 16) red[wv][16 + nn] = f1;  // combos 16,17
  }
  __syncthreads();

  const int tid = threadIdx.x;
  if (tid < NK) {
    const float f = red[0][tid] + red[1][tid] + red[2][tid] + red[3][tid];
    const int ai = anum[bn];
    const int elems[NE] = {1, 6, 8};
#pragma unroll
    for (int e = 0; e < NE; ++e) {
      const float g = (ai == elems[e]) ? 1.f : 0.f;   // one-hot z_i
      out[(bn*NE + e)*NK + tid] = g * f;
    }
  }
}

extern "C" void kernel_launch(void* const* d_in, const int* in_sizes, int n_in,
                              void* d_out, int out_size, void* d_ws, size_t ws_size,
                              hipStream_t stream) {
  (void)in_sizes; (void)n_in; (void)out_size; (void)d_ws; (void)ws_size;
  const float* pos  = (const float*)d_in[0];
  const float* cel  = (const float*)d_in[1];
  const int*   nbr  = (const int*)  d_in[2];
  const int*   nbrj = (const int*)  d_in[3];
  const int*   nbrk = (const int*)  d_in[4];
  const float* msk  = (const float*)d_in[5];
  const float* mskt = (const float*)d_in[6];
  const float* off  = (const float*)d_in[7];
  const float* offj = (const float*)d_in[8];
  const float* offk = (const float*)d_in[9];
  const int*   anum = (const int*)  d_in[10];
  float* out = (float*)d_out;

  dim3 grid(NB * NA), block(128);
  zernike_wmma_kernel<<<grid, block, 0, stream>>>(
      pos, cel, nbr, nbrj, nbrk, msk, mskt, off, offj, offk, anum, out);
}